// AudioSSCPConvBlock_88046829568093
// MI455X (gfx1250) — compile-verified
//
#include <hip/hip_runtime.h>
#include <hip/hip_bf16.h>

// Shapes (from the reference):
//   x      : [4, 1, 4096, 128] f32
//   conv_w : [128, 1, 3, 3]    f32   (OIHW)
//   scale  : [128]             f32
//   out    : [4, 128, 2048, 64] f32
#define B_   4
#define HIN_ 4096
#define WIN_ 128
#define CO_  128
#define T_   2048
#define F_   64
#define FC_  (F_ * CO_)   // 8192 elements per (b,t) group slice
#define EPS_ 1e-3f

#define XROW_ 132                  // padded LDS row stride (cols 0..128 used)
#define NXS_  (4 * XROW_)          // 3 input rows + 1 zero row (taps 9..11)
#define NWL_  (CO_ * 12)           // weights padded to 12 taps per channel

typedef float v2f __attribute__((ext_vector_type(2)));
typedef float v8f __attribute__((ext_vector_type(8)));

// ---------------------------------------------------------------------------
// Cooperative stage of one (b,t) tile into LDS, fully zero-padded so the hot
// loop needs no bounds checks:
//   xs[r][col], r=0..2 -> input rows 2t-1..2t+1 (H pad (1,1)); cols >=128
//   are the W right-pad; r=3 is an all-zero row used by padding taps 9..11.
//   wl[c][k], k=0..11 -> conv weights, taps 9..11 zero.
// ---------------------------------------------------------------------------
__device__ __forceinline__ void stage_tile(const float* __restrict__ x,
                                           const float* __restrict__ w,
                                           int b, int t,
                                           float* xs, float* wl) {
  const int tid = threadIdx.x;
  for (int i = tid; i < NXS_; i += 256) {
    const int r   = i / XROW_;
    const int col = i - r * XROW_;
    const int h   = 2 * t - 1 + r;
    float v = 0.0f;
    if (r < 3 && (unsigned)h < (unsigned)HIN_ && col < WIN_)
      v = x[((size_t)b * HIN_ + (unsigned)h) * WIN_ + col];
    xs[i] = v;
  }
  for (int i = tid; i < NWL_; i += 256) {
    const int c = i / 12;
    const int k = i - c * 12;
    wl[i] = (k < 9) ? w[c * 9 + k] : 0.0f;
  }
  __syncthreads();
}

// ---------------------------------------------------------------------------
// Per-wave loop-invariant fragments for D = A[16x12] * B[12x16]:
//   A layout (ISA 32-bit A 16x4): lanes 0-15 = M, v[0]=K0, v[1]=K1;
//                                 lanes 16-31 = M, v[0]=K2, v[1]=K3.
//   B mirrored with N in lanes. off0/off1 = LDS row/col offset of the input
//   element for taps k0/k1 (tap k -> row k/3 of xs, col k%3; taps 9..11 land
//   in the zero row).
// ---------------------------------------------------------------------------
__device__ __forceinline__ void make_frags(const float* wl, int c0, int lane,
                                           v2f* afrag, int* off0, int* off1) {
  const int half = lane >> 4;
  const int l    = lane & 15;
#pragma unroll
  for (int kk = 0; kk < 3; ++kk) {
    const int k0 = kk * 4 + half * 2;
    const int k1 = k0 + 1;
    afrag[kk].x = wl[(c0 + l) * 12 + k0];
    afrag[kk].y = wl[(c0 + l) * 12 + k1];
    const int kh0 = k0 / 3, kw0 = k0 - 3 * kh0;
    const int kh1 = k1 / 3, kw1 = k1 - 3 * kh1;
    off0[kk] = kh0 * XROW_ + kw0;
    off1[kk] = kh1 * XROW_ + kw1;
  }
}

// One 16(ch) x 16(freq) conv tile: 3 accumulating V_WMMA_F32_16X16X4_F32.
// fcol = 2*(f0 + lane%16) = input column of this lane's output position.
__device__ __forceinline__ v8f tile_wmma(const float* xs, const v2f* afrag,
                                         const int* off0, const int* off1,
                                         int fcol) {
  v8f acc = {};
#pragma unroll
  for (int kk = 0; kk < 3; ++kk) {
    v2f bb;
    bb.x = xs[off0[kk] + fcol];
    bb.y = xs[off1[kk] + fcol];
#if __has_builtin(__builtin_amdgcn_wmma_f32_16x16x4_f32)
    // (neg_a, A, neg_b, B, c_mod, C, reuse_a, reuse_b)
    acc = __builtin_amdgcn_wmma_f32_16x16x4_f32(
        false, afrag[kk], false, bb, (short)0, acc, false, false);
#else
    acc[0] = __builtin_fmaf(afrag[kk].x, bb.x,
                            __builtin_fmaf(afrag[kk].y, bb.y, acc[0]));
#endif
  }
  return acc;
}

// ---------------------------------------------------------------------------
// K1: per-(b,t) conv + deterministic block reduction of sum / sum-of-squares
// over all F_*CO_ = 8192 conv outputs of that row.
// grid = (T_, B_), block = 256 (8 waves: wave -> 16-channel block, 4 freq
// blocks per wave).
// ---------------------------------------------------------------------------
__global__ void __launch_bounds__(256)
k_conv_stats(const float* __restrict__ x, const float* __restrict__ w,
             float* __restrict__ S, float* __restrict__ Q) {
  const int t    = blockIdx.x;
  const int b    = blockIdx.y;
  const int lane = threadIdx.x & 31;
  const int wave = threadIdx.x >> 5;
  const int c0   = wave * 16;
  const int l    = lane & 15;

  __shared__ float xs[NXS_];
  __shared__ float wl[NWL_];

  // Warm L2 for the next t-row's input (gfx1250 global_prefetch_b8).
  if (threadIdx.x == 0 && t + 1 < T_) {
    __builtin_prefetch(&x[((size_t)b * HIN_ + (2 * t + 1)) * WIN_], 0, 1);
    __builtin_prefetch(&x[((size_t)b * HIN_ + (2 * t + 2)) * WIN_], 0, 1);
  }
  stage_tile(x, w, b, t, xs, wl);

  v2f afrag[3];
  int off0[3], off1[3];
  make_frags(wl, c0, lane, afrag, off0, off1);

  float s = 0.0f, q = 0.0f;
#pragma unroll
  for (int fb = 0; fb < 4; ++fb) {
    v8f d = tile_wmma(xs, afrag, off0, off1, 2 * (fb * 16 + l));
#pragma unroll
    for (int r = 0; r < 8; ++r) {
      float v = d[r];
      s += v;
      q += v * v;
    }
  }

  __shared__ float ls[256];
  __shared__ float lq[256];
  ls[threadIdx.x] = s;
  lq[threadIdx.x] = q;
  __syncthreads();
#pragma unroll
  for (int off = 128; off > 0; off >>= 1) {
    if ((int)threadIdx.x < off) {
      ls[threadIdx.x] += ls[threadIdx.x + off];
      lq[threadIdx.x] += lq[threadIdx.x + off];
    }
    __syncthreads();
  }
  if (threadIdx.x == 0) {
    S[b * T_ + t] = ls[0];
    Q[b * T_ + t] = lq[0];
  }
}

// ---------------------------------------------------------------------------
// Deterministic inclusive scan of buf[2048] with 256 threads (8 elems/thread
// serial + Hillis-Steele over per-thread totals in LDS).
// ---------------------------------------------------------------------------
__device__ __forceinline__ void block_scan2048(float* buf, float* tmp) {
  const int tid = threadIdx.x;
  float loc[8];
  float run = 0.0f;
#pragma unroll
  for (int i = 0; i < 8; ++i) {
    run += buf[tid * 8 + i];
    loc[i] = run;
  }
  tmp[tid] = run;
  __syncthreads();
  for (int off = 1; off < 256; off <<= 1) {
    float add = (tid >= off) ? tmp[tid - off] : 0.0f;
    __syncthreads();
    tmp[tid] += add;
    __syncthreads();
  }
  const float base = (tid > 0) ? tmp[tid - 1] : 0.0f;
#pragma unroll
  for (int i = 0; i < 8; ++i) buf[tid * 8 + i] = loc[i] + base;
  __syncthreads();
}

// ---------------------------------------------------------------------------
// K2: per-batch cumulative norm statistics.
//   cumS_t  -> mean_t = cumS_t / ((t+1)*FC_)
//   sq_t    = Q_t - 2*mean_t*S_t + FC_*mean_t^2
//   cumSq_t -> inv_t = rsqrt(cumSq_t / ((t+1)*FC_) + EPS)
// grid = B_, block = 256.
// ---------------------------------------------------------------------------
__global__ void __launch_bounds__(256)
k_scan(const float* __restrict__ S, const float* __restrict__ Q,
       float* __restrict__ mean, float* __restrict__ inv) {
  const int b   = blockIdx.x;
  const int tid = threadIdx.x;
  __shared__ float buf[T_];
  __shared__ float meanS[T_];
  __shared__ float tmp[256];

  for (int i = tid; i < T_; i += 256) buf[i] = S[b * T_ + i];
  __syncthreads();
  block_scan2048(buf, tmp);

  for (int i = tid; i < T_; i += 256) {
    float m = buf[i] / ((float)(i + 1) * (float)FC_);
    meanS[i] = m;
    mean[b * T_ + i] = m;
  }
  __syncthreads();

  for (int i = tid; i < T_; i += 256) {
    float m = meanS[i];
    float s = S[b * T_ + i];
    buf[i] = Q[b * T_ + i] - 2.0f * m * s + (float)FC_ * m * m;
  }
  __syncthreads();
  block_scan2048(buf, tmp);

  for (int i = tid; i < T_; i += 256) {
    float var = buf[i] / ((float)(i + 1) * (float)FC_);
    inv[b * T_ + i] = rsqrtf(var + EPS_);
  }
}

// ---------------------------------------------------------------------------
// K3: recompute conv tile from the (L2-resident) input -- cheaper than
// spilling the 256MB intermediate to HBM -- then normalize, scale per
// channel, ReLU, and store in [B, C, T, F] order. Lanes 0-15 map to
// consecutive f -> coalesced stores.
// ---------------------------------------------------------------------------
__global__ void __launch_bounds__(256)
k_norm_out(const float* __restrict__ x, const float* __restrict__ w,
           const float* __restrict__ scale,
           const float* __restrict__ mean, const float* __restrict__ inv,
           float* __restrict__ out) {
  const int t    = blockIdx.x;
  const int b    = blockIdx.y;
  const int lane = threadIdx.x & 31;
  const int wave = threadIdx.x >> 5;
  const int c0   = wave * 16;
  const int l    = lane & 15;
  const int half = lane >> 4;

  __shared__ float xs[NXS_];
  __shared__ float wl[NWL_];
  stage_tile(x, w, b, t, xs, wl);

  v2f afrag[3];
  int off0[3], off1[3];
  make_frags(wl, c0, lane, afrag, off0, off1);

  const float m  = mean[b * T_ + t];
  const float iv = inv[b * T_ + t];

  // Per-lane channel scales (loop-invariant across freq tiles).
  float sc[8];
#pragma unroll
  for (int r = 0; r < 8; ++r) sc[r] = scale[c0 + r + half * 8];

#pragma unroll
  for (int fb = 0; fb < 4; ++fb) {
    v8f d = tile_wmma(xs, afrag, off0, off1, 2 * (fb * 16 + l));
    const int f = fb * 16 + l;
#pragma unroll
    for (int r = 0; r < 8; ++r) {
      const int c = c0 + r + half * 8;   // D layout: VGPR r -> M=r / M=r+8
      float v = (d[r] - m) * iv * sc[r];
      v = fmaxf(v, 0.0f);                // ReLU
      out[(((size_t)b * CO_ + c) * T_ + t) * F_ + f] = v;
    }
  }
}

// ---------------------------------------------------------------------------
extern "C" void kernel_launch(void* const* d_in, const int* in_sizes, int n_in,
                              void* d_out, int out_size, void* d_ws, size_t ws_size,
                              hipStream_t stream) {
  const float* x     = (const float*)d_in[0];  // [4,1,4096,128]
  const float* w     = (const float*)d_in[1];  // [128,1,3,3]
  const float* scale = (const float*)d_in[2];  // [128]
  float*       out   = (float*)d_out;          // [4,128,2048,64]

  // Workspace layout: S, Q, mean, inv -- each B_*T_ floats (128 KB total).
  float* S    = (float*)d_ws;
  float* Q    = S + (size_t)B_ * T_;
  float* mean = Q + (size_t)B_ * T_;
  float* inv  = mean + (size_t)B_ * T_;

  dim3 grid(T_, B_);
  k_conv_stats<<<grid, 256, 0, stream>>>(x, w, S, Q);
  k_scan<<<B_, 256, 0, stream>>>(S, Q, mean, inv);
  k_norm_out<<<grid, 256, 0, stream>>>(x, w, scale, mean, inv, out);
}